// SimpleFA_82910048682189
// MI455X (gfx1250) — compile-verified
//
#include <hip/hip_runtime.h>
#include <stdint.h>

// Memory-bound slot-conditioned affine: out = alpha[slot,c]*x + beta[slot,c].
// ~410 MB mandatory HBM traffic -> ~18us @ 23.3 TB/s. Strategy:
//  - TDM tensor_load_to_lds stages alpha/beta channel-chunk columns into LDS,
//    using TDM padding (1 DW per 16 DW) for an odd LDS row stride (17) so the
//    random per-lane slot gathers spread across all 64 LDS banks.
//  - s_wait_tensorcnt + barrier, then streamed b128 NT loads/stores of x/out.
//  - slots loaded once per block (int4) and reused across 16 channels.

namespace {
constexpr int kB  = 16;
constexpr int kC  = 256;
constexpr int kH  = 112;
constexpr int kW  = 112;
constexpr int kS  = 256;              // slots
constexpr int kHW = kH * kW;          // 12544
constexpr int kCH = 16;               // channels staged per block
constexpr int kPixChunks = 7;         // pixel chunks per plane (12544 = 7*1792)
constexpr int kPix = kHW / kPixChunks;   // 1792 pixels per block
constexpr int kV4  = kPix / 4;           // 448 float4 groups per block
constexpr int kRow = kCH + 1;         // 17 DW padded LDS row stride

typedef float        f4    __attribute__((ext_vector_type(4)));
typedef int          i4    __attribute__((ext_vector_type(4)));
typedef unsigned int u32x4 __attribute__((ext_vector_type(4)));
typedef int          i32x8 __attribute__((ext_vector_type(8)));
typedef int          i32x4 __attribute__((ext_vector_type(4)));

// Issue one TDM 2D tile load: 256 rows (slots) x 16 elements (channels),
// row stride 256 elements in memory, padded to 17-DWORD rows in LDS.
__device__ __forceinline__ void tdm_load_table(const float* gsrc, uint32_t lds_off) {
  const uint64_t ga = (uint64_t)(uintptr_t)gsrc;          // byte address of tile start
  u32x4 g0;
  g0[0] = 1u;                                             // count=1 valid descriptor
  g0[1] = lds_off;                                        // lds_addr (bytes)
  g0[2] = (uint32_t)(ga & 0xffffffffull);                 // global_addr[31:0]
  g0[3] = (uint32_t)((ga >> 32) & 0x1ffffffull)           // global_addr[56:32]
        | (2u << 30);                                     // type = 2 ("image")
  i32x8 g1;
  g1[0] = (2 << 16)      // data_size = 4 bytes
        | (1 << 20)      // pad_enable
        | (3 << 22)      // pad_interval code 3 = 16 DWORDs
        | (0 << 25);     // pad_amount  code 0 = 1 DWORD  -> LDS row stride 17 DW
  g1[1] = (kC & 0xffff) << 16;                 // tensor_dim0 = 256 (low 16 bits)
  g1[2] = (kC >> 16) | ((kS & 0xffff) << 16);  // tensor_dim0 hi | tensor_dim1 lo
  g1[3] = (kS >> 16) | (kCH << 16);            // tensor_dim1 hi | tile_dim0 = 16
  g1[4] = kS;                                  // tile_dim1 = 256 rows, tile_dim2 = 0
  g1[5] = kC;                                  // tensor_dim0_stride = 256 elements
  g1[6] = 0;                                   // stride hi | tensor_dim1_stride lo
  g1[7] = 0;
  i32x4 gz4 = {0, 0, 0, 0};                    // groups 2/3 unused (2D tensor)
  i32x8 gz8 = {0, 0, 0, 0, 0, 0, 0, 0};        // extra group (clang-23 form), unused
  __builtin_amdgcn_tensor_load_to_lds(g0, g1, gz4, gz4, gz8, 0);
}
} // namespace

__global__ __launch_bounds__(256)
void slot_affine_tdm_kernel(const float* __restrict__ x,
                            const float* __restrict__ alpha,
                            const float* __restrict__ beta,
                            const int*   __restrict__ slot,
                            float* __restrict__ out) {
  __shared__ float ldsA[kS * kRow];   // 17408 B
  __shared__ float ldsB[kS * kRow];   // 17408 B

  const int b   = blockIdx.z;
  const int c0  = blockIdx.y * kCH;
  const int p0  = blockIdx.x * kPix;
  const int tid = threadIdx.x;

  // One wave issues both DMA tile loads (TDM ignores EXEC; issue once).
  if (tid == 0) {
    tdm_load_table(alpha + c0, (uint32_t)(uintptr_t)(void*)ldsA);
    tdm_load_table(beta  + c0, (uint32_t)(uintptr_t)(void*)ldsB);
    __builtin_amdgcn_s_wait_tensorcnt(0);
  }
  __syncthreads();

  const i4* sp = (const i4*)(slot + (long)b * kHW + p0);
  const long base = ((long)b * kC + c0) * (long)kHW + p0;
  const float* xb = x   + base;
  float*       ob = out + base;

  for (int v = tid; v < kV4; v += 256) {
    const i4 sv = sp[v];                       // 4 pixel slot ids, reused for 16 channels
    const int r0 = sv[0] * kRow;
    const int r1 = sv[1] * kRow;
    const int r2 = sv[2] * kRow;
    const int r3 = sv[3] * kRow;
    const float* xc = xb + 4 * v;
    float*       oc = ob + 4 * v;
#pragma unroll 4
    for (int cc = 0; cc < kCH; ++cc) {
      f4 xv = __builtin_nontemporal_load((const f4*)xc);
      f4 av = { ldsA[r0 + cc], ldsA[r1 + cc], ldsA[r2 + cc], ldsA[r3 + cc] };
      f4 bv = { ldsB[r0 + cc], ldsB[r1 + cc], ldsB[r2 + cc], ldsB[r3 + cc] };
      f4 ov;
      ov[0] = __builtin_fmaf(av[0], xv[0], bv[0]);
      ov[1] = __builtin_fmaf(av[1], xv[1], bv[1]);
      ov[2] = __builtin_fmaf(av[2], xv[2], bv[2]);
      ov[3] = __builtin_fmaf(av[3], xv[3], bv[3]);
      __builtin_nontemporal_store(ov, (f4*)oc);
      xc += kHW;
      oc += kHW;
    }
  }
}

extern "C" void kernel_launch(void* const* d_in, const int* in_sizes, int n_in,
                              void* d_out, int out_size, void* d_ws, size_t ws_size,
                              hipStream_t stream) {
  (void)in_sizes; (void)n_in; (void)out_size; (void)d_ws; (void)ws_size;
  const float* x     = (const float*)d_in[0];
  const float* alpha = (const float*)d_in[1];
  const float* beta  = (const float*)d_in[2];
  const int*   slot  = (const int*)d_in[3];
  float*       out   = (float*)d_out;

  dim3 grid(kPixChunks, kC / kCH, kB);   // 7 x 16 x 16 = 1792 blocks
  slot_affine_tdm_kernel<<<grid, 256, 0, stream>>>(x, alpha, beta, slot, out);
}